// ToIHeadTemplate_10307921511153
// MI455X (gfx1250) — compile-verified
//
#include <hip/hip_runtime.h>
#include <hip/hip_bf16.h>
#include <cstdint>

// ---------------------------------------------------------------------------
// Problem constants (match reference)
// ---------------------------------------------------------------------------
#define F_FRAMES 4
#define B_BATCH  4
#define H_DIM    376
#define W_DIM    376
#define A_ANCH   2
#define G_TRK    200
#define HWA      (H_DIM * W_DIM * A_ANCH)      // 282752
#define PC_RANGE 75.2f                          // -pc_range_x == -pc_range_y
#define INV_CELL 2.5f                           // 1 / (0.05 * 8)

#define TPB        256
#define ROW_F      (F_FRAMES * 7)               // 28 floats per output row
#define TILE_FLOATS (TPB * ROW_F)               // 7168 floats (28 KB LDS)

#define CPOL_STORE_NT 1                         // CPol TH[2:0]=1 -> TH_STORE_NT

// Native vector types (usable with nontemporal builtins on host & device).
typedef int   v4i __attribute__((vector_size(16)));
typedef float v4f __attribute__((vector_size(16)));
typedef v4i __attribute__((address_space(1))) g_v4i;   // global (device) AS
typedef v4i __attribute__((address_space(3))) l_v4i;   // LDS AS

// CK-style addrspace casts (LDS flat addr low 32 bits == LDS offset, ISA 10.2)
#define AS1_B128(p) ((g_v4i*)(uintptr_t)(p))
#define AS3_B128(p) ((l_v4i*)(uint32_t)(uintptr_t)(p))

// ---------------------------------------------------------------------------
// Kernel 1: dependent-gather tracklet assembly. One thread per (b,h,w,a) row.
//   frame 0  : streamed (non-temporal) 9-float read, coalesced
//   frame i>0: gather at BEV cell predicted from frame i-1's (pos + motion);
//              default (RT) caching so the hot frames stay resident in 192MB L2
//   output   : staged in LDS, drained via CDNA5 async LDS->global b128 stores
//              with TH_STORE_NT so the write-once stream doesn't evict frames
// ---------------------------------------------------------------------------
__global__ __launch_bounds__(TPB) void assemble_tracklets_kernel(
    const float* __restrict__ box_preds,   // [F,B,H,W,A,9]
    float*       __restrict__ out)         // [B, H*W*A, 28]
{
    __shared__ float tile[TILE_FLOATS];

    const int tid = threadIdx.x;
    const int t   = blockIdx.x * TPB + tid;     // row id in [0, B*HWA), exact grid
    const int b   = t / HWA;
    const int n   = t - b * HWA;                // row within batch: ((h*W)+w)*A + a
    const int aa  = n & (A_ANCH - 1);

    // ---- frame 0: read-once stream; NT keeps gather-hot frames 1..3 in L2 ----
    const float* p = box_preds + (size_t)(b * HWA + n) * 9;
    float box[9];
    #pragma unroll
    for (int j = 0; j < 9; ++j) box[j] = __builtin_nontemporal_load(p + j);

    float* trow = &tile[tid * ROW_F];
    #pragma unroll
    for (int j = 0; j < 7; ++j) trow[j] = box[j];

    // ---- frames 1..3: dependent gathers ----
    #pragma unroll
    for (int i = 1; i < F_FRAMES; ++i) {
        const float px = box[0] + box[7];
        const float py = box[1] + box[8];
        const float fx = floorf((px + PC_RANGE) * INV_CELL);
        const float fy = floorf((py + PC_RANGE) * INV_CELL);
        const int cx = (int)fminf(fmaxf(fx, 0.0f), (float)(W_DIM - 1));
        const int cy = (int)fminf(fmaxf(fy, 0.0f), (float)(H_DIM - 1));
        const float* q = box_preds +
            ((size_t)(i * B_BATCH + b) * HWA +
             (size_t)((cy * W_DIM + cx) * A_ANCH + aa)) * 9;
        #pragma unroll
        for (int j = 0; j < 9; ++j) box[j] = q[j];
        #pragma unroll
        for (int j = 0; j < 7; ++j) trow[i * 7 + j] = box[j];
    }

    __syncthreads();

    // ---- coalesced tile drain: 7 x 512B-per-wave contiguous b128 stores ----
    float* outBase = out + (size_t)blockIdx.x * TILE_FLOATS;
    #pragma unroll
    for (int it = 0; it < TILE_FLOATS / (TPB * 4); ++it) {   // 7 iterations
        const int k4 = (it * TPB + tid) * 4;                 // float index in tile
#if __has_builtin(__builtin_amdgcn_global_store_async_from_lds_b128)
        __builtin_amdgcn_global_store_async_from_lds_b128(
            AS1_B128(outBase + k4), AS3_B128(&tile[k4]),
            /*offset=*/0, /*cpol=*/CPOL_STORE_NT);
#else
        __builtin_nontemporal_store(*(const v4f*)&tile[k4],
                                    (v4f*)(outBase + k4));
#endif
    }

#if __has_builtin(__builtin_amdgcn_s_wait_asynccnt)
    __builtin_amdgcn_s_wait_asynccnt(0);
#else
    asm volatile("s_wait_asynccnt 0" ::: "memory");
#endif
}

// ---------------------------------------------------------------------------
// Kernel 2: gt_tracklets rearrange [B,G,20] -> [B,G,29]  +  mask [B,G,4]
// ---------------------------------------------------------------------------
__global__ __launch_bounds__(TPB) void gt_rearrange_kernel(
    const float* __restrict__ gt,        // [B,G,20]
    const int*   __restrict__ nbb,       // [B,G,1]
    float*       __restrict__ out_gt,    // [B,G,29]
    float*       __restrict__ out_mask)  // [B,G,4]
{
    const int r = blockIdx.x * blockDim.x + threadIdx.x;
    if (r >= B_BATCH * G_TRK) return;

    const float* g = gt + (size_t)r * 20;
    float*       o = out_gt + (size_t)r * 29;

    float v[29];
    #pragma unroll
    for (int j = 0; j < 7; ++j) v[j] = g[j];

    #pragma unroll
    for (int i = 1; i < F_FRAMES; ++i) {
        const int base = 7 + (i - 1) * 4;
        v[i * 7 + 0] = g[base + 0];   // x of frame -i
        v[i * 7 + 1] = g[base + 1];   // y
        v[i * 7 + 2] = g[base + 2];   // z
        v[i * 7 + 3] = g[3];          // shared l
        v[i * 7 + 4] = g[4];          // shared w
        v[i * 7 + 5] = g[5];          // shared h
        v[i * 7 + 6] = g[base + 3];   // heading of frame -i
    }
    v[28] = g[19];                    // trailing label/score

    #pragma unroll
    for (int j = 0; j < 29; ++j) __builtin_nontemporal_store(v[j], o + j);

    const int nb = nbb[r];
    #pragma unroll
    for (int i = 0; i < F_FRAMES; ++i)
        __builtin_nontemporal_store((nb >= i + 1) ? 1.0f : 0.0f,
                                    out_mask + r * F_FRAMES + i);
}

// ---------------------------------------------------------------------------
// Launch
// ---------------------------------------------------------------------------
extern "C" void kernel_launch(void* const* d_in, const int* in_sizes, int n_in,
                              void* d_out, int out_size, void* d_ws, size_t ws_size,
                              hipStream_t stream) {
    const float* box_preds = (const float*)d_in[0];   // [4,4,376,376,2,9] f32
    const float* gt        = (const float*)d_in[1];   // [4,200,20] f32
    const int*   nbb       = (const int*)d_in[2];     // [4,200,1] i32

    float* out      = (float*)d_out;
    float* out_gt   = out + (size_t)B_BATCH * HWA * ROW_F;        // +31,668,224
    float* out_mask = out_gt + (size_t)B_BATCH * G_TRK * 29;      // +23,200

    // 1,131,008 rows == 4418 blocks * 256 threads exactly (no tail)
    assemble_tracklets_kernel<<<(B_BATCH * HWA) / TPB, TPB, 0, stream>>>(
        box_preds, out);

    gt_rearrange_kernel<<<(B_BATCH * G_TRK + TPB - 1) / TPB, TPB, 0, stream>>>(
        gt, nbb, out_gt, out_mask);
}